// NadeMaskLayer_10934986736318
// MI455X (gfx1250) — compile-verified
//
#include <hip/hip_runtime.h>
#include <stdint.h>

// NADE mask layer for MI455X (gfx1250, wave32).
// One workgroup (512 threads = 16 waves) per row:
//   1. wave0 kicks a TDM tensor_load_to_lds DMA of the 16KB x-row (overlaps 2+3)
//   2. all waves generate threefry2x32 score bits, build 35-bit composite keys
//   3. stable in-LDS bitonic sort of 4096 keys (78 passes)
//   4. s_wait_tensorcnt + barrier, then fused (x*mask, mask) b128 stores
#define TILE_D 4096u
#define NT     512u

__device__ __forceinline__ unsigned rotl32(unsigned v, int r) {
  return (v << r) | (v >> (32 - r));
}

// Exact JAX threefry2x32 (20 rounds).
__device__ __forceinline__ void threefry2x32(unsigned k0, unsigned k1,
                                             unsigned c0, unsigned c1,
                                             unsigned& o0, unsigned& o1) {
  const unsigned ks2 = k0 ^ k1 ^ 0x1BD11BDAu;
  unsigned x0 = c0 + k0;
  unsigned x1 = c1 + k1;
#define TF_RND(r) { x0 += x1; x1 = rotl32(x1, (r)); x1 ^= x0; }
  TF_RND(13) TF_RND(15) TF_RND(26) TF_RND(6)
  x0 += k1;  x1 += ks2 + 1u;
  TF_RND(17) TF_RND(29) TF_RND(16) TF_RND(24)
  x0 += ks2; x1 += k0 + 2u;
  TF_RND(13) TF_RND(15) TF_RND(26) TF_RND(6)
  x0 += k0;  x1 += k1 + 3u;
  TF_RND(17) TF_RND(29) TF_RND(16) TF_RND(24)
  x0 += k1;  x1 += ks2 + 4u;
  TF_RND(13) TF_RND(15) TF_RND(26) TF_RND(6)
  x0 += ks2; x1 += k0 + 5u;
#undef TF_RND
  o0 = x0; o1 = x1;
}

#if __has_builtin(__builtin_amdgcn_tensor_load_to_lds) && \
    __has_builtin(__builtin_amdgcn_s_wait_tensorcnt)
#define NADE_USE_TDM 1
#else
#define NADE_USE_TDM 0
#endif

__global__ __launch_bounds__(NT) void nade_mask_kernel(
    const float* __restrict__ x, float* __restrict__ out,
    unsigned Bn, unsigned halfN) {
  __shared__ unsigned long long keys[TILE_D];   // 32 KB: composite sort keys
#if NADE_USE_TDM
  __shared__ float xrow[TILE_D];                // 16 KB: TDM-staged x row
#endif
  __shared__ unsigned sh_n;

  const unsigned tid = threadIdx.x;
  const unsigned b = blockIdx.x;
  const size_t row_in = (size_t)b * TILE_D;
  const size_t row_out = (size_t)b * (2u * TILE_D);

#if NADE_USE_TDM
  // ---- Tensor Data Mover: async DMA x[b, :] -> LDS (wave 0 issues once) ----
  if ((tid >> 5) == 0u) {
    typedef unsigned int u32x4 __attribute__((ext_vector_type(4)));
    typedef int i32x4 __attribute__((ext_vector_type(4)));
    typedef int i32x8 __attribute__((ext_vector_type(8)));
    const unsigned long long ga = (unsigned long long)(const void*)(x + row_in);
    const unsigned la = (unsigned)(unsigned long long)(void*)xrow;
    // D# group 0: count=1, lds_addr, global_addr[56:0], type=2 (bits 127:126 = 10b)
    u32x4 g0;
    g0.x = 0x1u;
    g0.y = la;
    g0.z = (unsigned)(ga & 0xFFFFFFFFull);
    g0.w = (unsigned)((ga >> 32) & 0x01FFFFFFull) | 0x80000000u;
    // D# group 1: data_size=2 (4B), tensor_dim0=4096, tensor_dim1=1,
    //             tile_dim0=4096 (1D tile), tensor_dim0_stride=4096
    i32x8 g1;
    g1[0] = 0x00020000;                       // wg_mask=0, data_size=2
    g1[1] = (int)((TILE_D & 0xFFFFu) << 16);  // tensor_dim0 low16 @ [63:48]
    g1[2] = 0x00010000;                       // tensor_dim0 hi=0, tensor_dim1=1
    g1[3] = (int)((TILE_D & 0xFFFFu) << 16);  // tile_dim0 @ [127:112]
    g1[4] = 0;                                // tile_dim1=0, tile_dim2=0 (unused)
    g1[5] = (int)TILE_D;                      // tensor_dim0_stride low32
    g1[6] = 0;
    g1[7] = 0;
    i32x4 gz = {0, 0, 0, 0};
#if defined(__clang_major__) && (__clang_major__ >= 23)
    i32x8 gz8 = {0, 0, 0, 0, 0, 0, 0, 0};
    __builtin_amdgcn_tensor_load_to_lds(g0, g1, gz, gz, gz8, 0);
#else
    __builtin_amdgcn_tensor_load_to_lds(g0, g1, gz, gz, 0);
#endif
  }
#else
  __builtin_prefetch(x + row_in + tid * (TILE_D / NT), 0, 0);
#endif

  // ---- Derive split keys: split(key(42)) via threefry on iota(4) ----
  unsigned sA0, sB0, sA1, sB1;
  threefry2x32(0u, 42u, 0u, 2u, sA0, sB0);
  threefry2x32(0u, 42u, 1u, 3u, sA1, sB1);
  const unsigned ki0 = sA0, ki1 = sA1;   // k_ints
  const unsigned kp0 = sB0, kp1 = sB1;   // k_perm

  // ---- n = randint(k_ints, 0, 4096)[b]; span=2^12 => lower_bits & 0xFFF ----
  if (tid == 0u) {
    unsigned o0, o1;
    threefry2x32(ki0, ki1, b, b + Bn, o0, o1);  // lower_bits[b] = word1
    sh_n = o1 & (TILE_D - 1u);
  }

  // ---- Generate uniform bits, build composite keys ((bits>>9)<<12 | j) ----
  constexpr unsigned EPT = TILE_D / NT;  // 8 elements per thread
  const unsigned j0 = tid * EPT;
#pragma unroll
  for (unsigned e = 0; e < EPT; ++e) {
    const unsigned j = j0 + e;
    const unsigned g = b * TILE_D + j;  // flat index < 2^26
    unsigned o0, o1, u;
    if (g < halfN) { threefry2x32(kp0, kp1, g, g + halfN, o0, o1); u = o0; }
    else           { threefry2x32(kp0, kp1, g - halfN, g, o0, o1); u = o1; }
    keys[j] = ((unsigned long long)(u >> 9) << 12) | (unsigned long long)j;
  }

  // ---- In-LDS bitonic sort, ascending, unique keys => exact stable argsort --
  for (unsigned k = 2u; k <= TILE_D; k <<= 1) {
    for (unsigned js = k >> 1; js > 0u; js >>= 1) {
      __syncthreads();
#pragma unroll
      for (unsigned t = tid; t < TILE_D / 2u; t += NT) {
        const unsigned i = 2u * t - (t & (js - 1u));
        const unsigned p = i + js;
        const bool up = ((i & k) == 0u);
        const unsigned long long av = keys[i];
        const unsigned long long bv = keys[p];
        if ((av > bv) == up) { keys[i] = bv; keys[p] = av; }
      }
    }
  }

#if NADE_USE_TDM
  if ((tid >> 5) == 0u) __builtin_amdgcn_s_wait_tensorcnt(0);
#endif
  __syncthreads();

  // ---- mask[r] = (orig_index(sorted r) < n); fused multiply + b128 stores ---
  const unsigned n = sh_n;
  float mv[EPT];
  float xv[EPT];
#pragma unroll
  for (unsigned e = 0; e < EPT; ++e) {
    const unsigned long long kk = keys[j0 + e];
    mv[e] = (((unsigned)kk & (TILE_D - 1u)) < n) ? 1.0f : 0.0f;
  }
#if NADE_USE_TDM
#pragma unroll
  for (unsigned e = 0; e < EPT; ++e) xv[e] = xrow[j0 + e];
#else
  {
    const float4* xg = (const float4*)(x + row_in + j0);
    const float4 xa = xg[0];
    const float4 xb = xg[1];
    xv[0] = xa.x; xv[1] = xa.y; xv[2] = xa.z; xv[3] = xa.w;
    xv[4] = xb.x; xv[5] = xb.y; xv[6] = xb.z; xv[7] = xb.w;
  }
#endif
  float4* oA = (float4*)(out + row_out + j0);
  oA[0] = make_float4(xv[0] * mv[0], xv[1] * mv[1], xv[2] * mv[2], xv[3] * mv[3]);
  oA[1] = make_float4(xv[4] * mv[4], xv[5] * mv[5], xv[6] * mv[6], xv[7] * mv[7]);
  float4* oM = (float4*)(out + row_out + TILE_D + j0);
  oM[0] = make_float4(mv[0], mv[1], mv[2], mv[3]);
  oM[1] = make_float4(mv[4], mv[5], mv[6], mv[7]);
}

extern "C" void kernel_launch(void* const* d_in, const int* in_sizes, int n_in,
                              void* d_out, int out_size, void* d_ws, size_t ws_size,
                              hipStream_t stream) {
  (void)n_in; (void)out_size; (void)d_ws; (void)ws_size;
  const float* x = (const float*)d_in[0];
  float* out = (float*)d_out;
  const unsigned Bn = (unsigned)(in_sizes[0] / (int)TILE_D);        // 16384
  const unsigned halfN = (unsigned)(((unsigned long long)Bn * TILE_D) >> 1);  // 2^25
  nade_mask_kernel<<<dim3(Bn), dim3(NT), 0, stream>>>(x, out, Bn, halfN);
}